// ElevateH2toH3_36532991820312
// MI455X (gfx1250) — compile-verified
//
#include <hip/hip_runtime.h>
#include <math.h>

// Problem constants (match reference)
#define BB 4
#define SS 512
#define DD 256
#define LN_EPS 1e-5f

typedef __attribute__((ext_vector_type(2))) float v2f;
typedef __attribute__((ext_vector_type(8))) float v8f;

// ---------------------------------------------------------------------------
// 1) Row squared-norms: one wave per row (wave32), shuffle reduce.
// ---------------------------------------------------------------------------
__global__ void norms_kernel(const float* __restrict__ x, float* __restrict__ norms) {
  const int wave = (blockIdx.x * blockDim.x + threadIdx.x) >> 5;  // global wave id
  const int lane = threadIdx.x & 31;
  if (wave >= BB * SS) return;
  const float* row = x + (size_t)wave * DD;
  float s = 0.f;
  for (int d = lane; d < DD; d += 32) { float v = row[d]; s += v * v; }
  #pragma unroll
  for (int off = 16; off > 0; off >>= 1) s += __shfl_xor(s, off, 32);
  if (lane == 0) norms[wave] = s;
}

// ---------------------------------------------------------------------------
// 2) Gram matrix via fp32 WMMA: dists = sqrt(max(ni + nj - 2*Xi.Xj, 0))
//    One wave computes one 16x16 tile of G.  4096 tiles total, 8 waves/block.
//    A-fragment (16x4 f32): lane l holds row M=l%16, K = 2*(l/16)..2*(l/16)+1
//    B-fragment (4x16 f32): lane l holds col N=l%16, K = 2*(l/16)..2*(l/16)+1
//    -> both are one contiguous float2 load per lane per k-step.
//    C/D layout: VGPR v: lanes 0-15 -> (M=v, N=lane); lanes 16-31 -> (M=v+8).
// ---------------------------------------------------------------------------
__global__ void gram_kernel(const float* __restrict__ x,
                            const float* __restrict__ norms,
                            float* __restrict__ dists) {
  const int wave  = threadIdx.x >> 5;
  const int lane  = threadIdx.x & 31;
  const int tile  = blockIdx.x * 8 + wave;   // 4096 tiles total (exact grid)
  const int batch = tile >> 10;              // 1024 tiles per batch (32x32)
  const int t     = tile & 1023;
  const int tr    = t >> 5, tc = t & 31;
  const int m     = lane & 15, half = lane >> 4;

  const float* xa = x + (size_t)(batch * SS + tr * 16 + m) * DD;
  const float* xb = x + (size_t)(batch * SS + tc * 16 + m) * DD;

  v8f c = {};
  for (int k0 = 0; k0 < DD; k0 += 4) {
    v2f a = *(const v2f*)(xa + k0 + 2 * half);
    v2f b = *(const v2f*)(xb + k0 + 2 * half);
    // 8 args: (neg_a, A, neg_b, B, c_mod, C, reuse_a, reuse_b)
    c = __builtin_amdgcn_wmma_f32_16x16x4_f32(false, a, false, b, (short)0, c,
                                              false, false);
  }

  const int rbase = tr * 16, cbase = tc * 16;
  const float nj = norms[batch * SS + cbase + m];
  #pragma unroll
  for (int v = 0; v < 8; ++v) {
    const int mr = v + 8 * half;
    const float ni = norms[batch * SS + rbase + mr];
    float sq = ni + nj - 2.0f * c[v];
    float d  = (sq > 0.f) ? sqrtf(sq) : 0.f;   // subgradient-0 at zero, like ref
    dists[(size_t)(batch * SS + rbase + mr) * SS + cbase + m] = d;
  }
}

// ---------------------------------------------------------------------------
// 3) Lower median (rank k = (n-1)/2) via bitwise radix-select on float bits
//    (all dists >= 0 so uint order == float order). One block per batch.
// ---------------------------------------------------------------------------
__global__ void median_kernel(const float* __restrict__ dists,
                              float* __restrict__ thr) {
  __shared__ unsigned red[1024];
  __shared__ unsigned sprefix;
  const int b = blockIdx.x, tid = threadIdx.x;
  const unsigned* u = (const unsigned*)(dists + (size_t)b * SS * SS);
  const int n = SS * SS;
  const unsigned k = (unsigned)((n - 1) / 2);
  if (tid == 0) sprefix = 0u;
  __syncthreads();
  for (int bit = 30; bit >= 0; --bit) {
    const unsigned cand = sprefix | (1u << bit);
    unsigned cnt = 0;
    for (int i = tid; i < n; i += 1024) cnt += (u[i] < cand) ? 1u : 0u;
    red[tid] = cnt; __syncthreads();
    for (int s = 512; s > 0; s >>= 1) {
      if (tid < s) red[tid] += red[tid + s];
      __syncthreads();
    }
    if (tid == 0) { if (red[0] <= k) sprefix = cand; }  // kth >= cand
    __syncthreads();
  }
  if (tid == 0) thr[b] = __uint_as_float(sprefix);
}

// ---------------------------------------------------------------------------
// 4) Graph + triu distance statistics. One block (1024 thr) per batch.
// ---------------------------------------------------------------------------
__device__ __forceinline__ float blk_add(float v, float* red, int tid) {
  red[tid] = v; __syncthreads();
  for (int s = 512; s > 0; s >>= 1) { if (tid < s) red[tid] += red[tid + s]; __syncthreads(); }
  float r = red[0]; __syncthreads(); return r;
}
__device__ __forceinline__ float blk_min(float v, float* red, int tid) {
  red[tid] = v; __syncthreads();
  for (int s = 512; s > 0; s >>= 1) { if (tid < s) red[tid] = fminf(red[tid], red[tid + s]); __syncthreads(); }
  float r = red[0]; __syncthreads(); return r;
}
__device__ __forceinline__ float blk_max(float v, float* red, int tid) {
  red[tid] = v; __syncthreads();
  for (int s = 512; s > 0; s >>= 1) { if (tid < s) red[tid] = fmaxf(red[tid], red[tid + s]); __syncthreads(); }
  float r = red[0]; __syncthreads(); return r;
}

__global__ void stats_kernel(const float* __restrict__ dists,
                             const float* __restrict__ thr,
                             float* __restrict__ topo) {
  __shared__ float red[1024];
  const int b = blockIdx.x, tid = threadIdx.x;
  const float* dm = dists + (size_t)b * SS * SS;
  const float t = thr[b];
  float conn = 0.f, sum = 0.f, sumsq = 0.f, mn = INFINITY, mx = -INFINITY;
  for (int i = tid; i < SS * SS; i += 1024) {
    const int r = i >> 9, cix = i & (SS - 1);
    const float d = dm[i];
    if (d < t && r != cix) conn += 1.f;
    if (cix >= r) { sum += d; sumsq += d * d; mn = fminf(mn, d); mx = fmaxf(mx, d); }
  }
  conn  = blk_add(conn,  red, tid);
  sum   = blk_add(sum,   red, tid);
  sumsq = blk_add(sumsq, red, tid);
  mn    = blk_min(mn,    red, tid);
  mx    = blk_max(mx,    red, tid);
  if (tid == 0) {
    const float cnt  = (float)(SS * (SS + 1) / 2);
    const float mean = sum / cnt;
    float var = (sumsq - cnt * mean * mean) / (cnt - 1.f);
    float sd  = sqrtf(fmaxf(var, 0.f));
    topo[b * 8 + 0] = conn / (float)(SS * (SS - 1));
    topo[b * 8 + 2] = (conn / (float)SS) / (float)SS;  // avg_degree / S
    topo[b * 8 + 3] = mean;
    topo[b * 8 + 4] = sd;
    topo[b * 8 + 5] = mn;
    topo[b * 8 + 6] = mx;
    topo[b * 8 + 7] = 0.f;
  }
}

// ---------------------------------------------------------------------------
// 5) Connected components: Jacobi min-label propagation, labels in LDS.
//    One block (512 threads == 512 nodes) per batch.
// ---------------------------------------------------------------------------
__global__ void comp_kernel(const float* __restrict__ dists,
                            const float* __restrict__ thr,
                            float* __restrict__ topo) {
  __shared__ int lab[SS];
  __shared__ int changed;
  __shared__ int cnt;
  const int b = blockIdx.x, i = threadIdx.x;
  const float t = thr[b];
  const float* row = dists + ((size_t)b * SS + i) * SS;
  lab[i] = i;
  __syncthreads();
  for (int it = 0; it < SS; ++it) {
    if (i == 0) changed = 0;
    __syncthreads();
    int best = lab[i];
    for (int j = 0; j < SS; ++j) {
      if (j != i && row[j] < t) { const int lj = lab[j]; if (lj < best) best = lj; }
    }
    __syncthreads();
    if (best != lab[i]) { lab[i] = best; changed = 1; }
    __syncthreads();
    if (!changed) break;   // uniform across block
  }
  if (i == 0) cnt = 0;
  __syncthreads();
  if (lab[i] == i) atomicAdd(&cnt, 1);
  __syncthreads();
  if (i == 0) topo[b * 8 + 1] = (float)cnt / (float)SS;
}

// ---------------------------------------------------------------------------
// 6) Fused signal + LayerNorm. One block (256 thr = D) per (b, s) row.
// ---------------------------------------------------------------------------
__global__ void ln_kernel(const float* __restrict__ x, const float* __restrict__ W,
                          const float* __restrict__ bias,
                          const float* __restrict__ gamma,
                          const float* __restrict__ beta,
                          const float* __restrict__ topo,
                          float* __restrict__ out) {
  __shared__ float red[DD];
  __shared__ float tp[8];
  const int row = blockIdx.x;          // b*S + s
  const int b   = row / SS;
  const int d   = threadIdx.x;
  if (d < 8) tp[d] = topo[b * 8 + d];
  __syncthreads();
  float sig = bias[d];
  #pragma unroll
  for (int q = 0; q < 8; ++q) sig += tp[q] * W[d * 8 + q];
  const float h = x[(size_t)row * DD + d] + sig;

  red[d] = h; __syncthreads();
  for (int s = DD / 2; s > 0; s >>= 1) { if (d < s) red[d] += red[d + s]; __syncthreads(); }
  const float mu = red[0] / (float)DD;
  __syncthreads();
  const float dev = h - mu;
  red[d] = dev * dev; __syncthreads();
  for (int s = DD / 2; s > 0; s >>= 1) { if (d < s) red[d] += red[d + s]; __syncthreads(); }
  const float var = red[0] / (float)DD;
  out[(size_t)row * DD + d] = dev * rsqrtf(var + LN_EPS) * gamma[d] + beta[d];
}

// ---------------------------------------------------------------------------
extern "C" void kernel_launch(void* const* d_in, const int* in_sizes, int n_in,
                              void* d_out, int out_size, void* d_ws, size_t ws_size,
                              hipStream_t stream) {
  const float* x     = (const float*)d_in[0];  // [B,S,D]
  const float* W     = (const float*)d_in[1];  // [D,8]
  const float* bias  = (const float*)d_in[2];  // [D]
  const float* gamma = (const float*)d_in[3];  // [D]
  const float* beta  = (const float*)d_in[4];  // [D]
  float* out = (float*)d_out;

  // workspace layout
  float* dists = (float*)d_ws;                       // B*S*S floats (4 MB)
  float* norms = dists + (size_t)BB * SS * SS;       // B*S floats
  float* thr   = norms + (size_t)BB * SS;            // B floats
  float* topo  = thr + BB;                           // B*8 floats

  // 1) row norms: one wave/row, 8 waves/block -> 2048 rows / 8 = 256 blocks
  norms_kernel<<<(BB * SS) / 8, 256, 0, stream>>>(x, norms);
  // 2) Gram matrix + sqrt: 4096 tiles, 8 waves/block -> 512 blocks (exact)
  gram_kernel<<<512, 256, 0, stream>>>(x, norms, dists);
  // 3) per-batch median threshold
  median_kernel<<<BB, 1024, 0, stream>>>(dists, thr);
  // 4) per-batch graph/distance stats
  stats_kernel<<<BB, 1024, 0, stream>>>(dists, thr, topo);
  // 5) per-batch connected components
  comp_kernel<<<BB, SS, 0, stream>>>(dists, thr, topo);
  // 6) fused signal + LayerNorm
  ln_kernel<<<BB * SS, DD, 0, stream>>>(x, W, bias, gamma, beta, topo, out);
}